// UpsampleFeat_3925600108914
// MI455X (gfx1250) — compile-verified
//
#include <hip/hip_runtime.h>

typedef __attribute__((ext_vector_type(2))) float v2f;
typedef __attribute__((ext_vector_type(8))) float v8f;

#define B_ 8
#define N_ 8192
#define S_ 2048
#define C_ 256
#define K_ 8

// sorted ascending top-K register insert; no-op when d >= current max
__device__ __forceinline__ void insert_sorted(float (&bd)[K_], int (&bi)[K_],
                                              float d, int id) {
    #pragma unroll
    for (int q = 0; q < K_; ++q) {
        bool lt = d < bd[q];
        float td = bd[q]; int ti = bi[q];
        bd[q] = lt ? d : td;  bi[q] = lt ? id : ti;
        d = lt ? td : d;      id = lt ? ti : id;
    }
}

// scan 8 candidates with a single min-tree guard (one branch per packet)
__device__ __forceinline__ void scan8(float4 v0, float4 v1, int id0,
                                      float (&bd)[K_], int (&bi)[K_]) {
    float m = fminf(fminf(fminf(v0.x, v0.y), fminf(v0.z, v0.w)),
                    fminf(fminf(v1.x, v1.y), fminf(v1.z, v1.w)));
    if (m < bd[K_ - 1]) {
        float cand[8] = {v0.x, v0.y, v0.z, v0.w, v1.x, v1.y, v1.z, v1.w};
        #pragma unroll
        for (int j = 0; j < 8; ++j) insert_sorted(bd, bi, cand[j], id0 + j);
    }
}

// =====================================================================
// Kernel 1: KNN (K=8) via WMMA score-GEMM + inverse-distance weights.
// grid: (N_/128, B_), block: 256 = 8 waves, each wave owns 16 queries.
// Score D[m][s] = |s|^2 - 2 q_m . s  (row-constant |q|^2 dropped: ordering safe)
// Tile loop unrolled x2: two independent WMMAs overlap the LDS transpose.
// =====================================================================
__global__ __launch_bounds__(256)
void knn_weights_kernel(const float* __restrict__ xyz,   // [B,3,N]
                        const float* __restrict__ sxyz,  // [B,3,S]
                        int* __restrict__ idx_out,       // [B,N,8]
                        float* __restrict__ w_out)       // [B,N,8]
{
    __shared__ float4 pts[S_];          // (sx,sy,sz,|s|^2)       32 KB
    __shared__ float  tileAll[8 * 512]; // per-wave 2x 16x16 tile 16 KB
    __shared__ float  mrgD[8 * 256];    // merge scratch           8 KB
    __shared__ int    mrgI[8 * 256];    //                         8 KB

    const int b    = blockIdx.y;
    const int tid  = threadIdx.x;
    const int wave = tid >> 5;
    const int lane = tid & 31;
    const int hi   = lane >> 4;   // half-wave: K-pair select (A/B), row+8 (D)
    const int rw   = lane & 15;   // row owned (and column lane inside tile)

    for (int s = tid; s < S_; s += 256) {
        float sx = sxyz[(b * 3 + 0) * S_ + s];
        float sy = sxyz[(b * 3 + 1) * S_ + s];
        float sz = sxyz[(b * 3 + 2) * S_ + s];
        pts[s] = make_float4(sx, sy, sz, sx * sx + sy * sy + sz * sz);
    }
    __syncthreads();

    const int nq = blockIdx.x * 128 + wave * 16 + rw;

    // A fragment (16x4 f32): rows [-2x,-2y,-2z,1]; lanes0-15 K0,K1; 16-31 K2,K3
    float qx = xyz[(b * 3 + 0) * N_ + nq];
    float qy = xyz[(b * 3 + 1) * N_ + nq];
    float qz = xyz[(b * 3 + 2) * N_ + nq];
    v2f a;
    a.x = hi ? (-2.0f * qz) : (-2.0f * qx);
    a.y = hi ? 1.0f         : (-2.0f * qy);

    float bd[K_]; int bi[K_];
    #pragma unroll
    for (int j = 0; j < K_; ++j) { bd[j] = 3.0e38f; bi[j] = 0; }

    float* tile = &tileAll[wave * 512];

    for (int t = 0; t < S_ / 32; ++t) {
        const int col0 = t * 32;
        // B fragments (4x16 f32): lane = column; halves hold (K0,K1)/(K2,K3)
        float4 p0 = pts[col0 + rw];
        float4 p1 = pts[col0 + 16 + rw];
        v2f bf0, bf1;
        bf0.x = hi ? p0.z : p0.x;  bf0.y = hi ? p0.w : p0.y;
        bf1.x = hi ? p1.z : p1.x;  bf1.y = hi ? p1.w : p1.y;
        v8f c0 = {}, c1 = {};
        c0 = __builtin_amdgcn_wmma_f32_16x16x4_f32(false, a, false, bf0,
                                                   (short)0, c0, false, false);
        c1 = __builtin_amdgcn_wmma_f32_16x16x4_f32(false, a, false, bf1,
                                                   (short)0, c1, false, false);
        // D layout: VGPR r -> row r (lanes0-15) / r+8 (lanes16-31), col = rw.
        #pragma unroll
        for (int r = 0; r < 8; ++r) {
            tile[(r + 8 * hi) * 16 + rw]       = c0[r];
            tile[256 + (r + 8 * hi) * 16 + rw] = c1[r];
        }
        const float4* tv0 =
            reinterpret_cast<const float4*>(&tile[rw * 16 + hi * 8]);
        const float4* tv1 =
            reinterpret_cast<const float4*>(&tile[256 + rw * 16 + hi * 8]);
        float4 a0 = tv0[0], a1 = tv0[1], b0 = tv1[0], b1 = tv1[1];
        scan8(a0, a1, col0 + hi * 8,      bd, bi);
        scan8(b0, b1, col0 + 16 + hi * 8, bd, bi);
    }

    // merge the two half-wave top-8 lists per row through LDS
    #pragma unroll
    for (int j = 0; j < K_; ++j) {
        mrgD[wave * 256 + rw * 16 + hi * 8 + j] = bd[j];
        mrgI[wave * 256 + rw * 16 + hi * 8 + j] = bi[j];
    }
    __syncthreads();

    if (lane < 16) {
        float fd[K_]; int fi[K_];
        #pragma unroll
        for (int j = 0; j < K_; ++j) { fd[j] = 3.0e38f; fi[j] = 0; }
        #pragma unroll
        for (int j = 0; j < 16; ++j)
            insert_sorted(fd, fi, mrgD[wave * 256 + rw * 16 + j],
                                  mrgI[wave * 256 + rw * 16 + j]);

        float inv[K_]; float sum = 0.0f;
        #pragma unroll
        for (int k = 0; k < K_; ++k) {
            float4 p = pts[fi[k]];
            float dx = qx - p.x, dy = qy - p.y, dz = qz - p.z;
            float dist = sqrtf(dx * dx + dy * dy + dz * dz);
            dist = fmaxf(dist, 1e-10f);
            inv[k] = 1.0f / dist;
            sum += inv[k];
        }
        float rs = 1.0f / sum;
        #pragma unroll
        for (int k = 0; k < K_; ++k) {
            idx_out[((size_t)b * N_ + nq) * K_ + k] = fi[k];
            w_out  [((size_t)b * N_ + nq) * K_ + k] = inv[k] * rs;
        }
    }
}

// =====================================================================
// Kernel 2: weighted feature gather.  CH=8 feature rows (64 KB) staged
// into LDS with gfx1250 async-to-LDS copies (ASYNCcnt), then streamed.
// grid: (C_/CH, B_, NSPLIT), block: 256
// =====================================================================
#define CH 8
#define NSPLIT 4

__global__ __launch_bounds__(256)
void interp_kernel(const float* __restrict__ sf,      // [B,C,S]
                   const int* __restrict__ idx_in,    // [B,N,8]
                   const float* __restrict__ w_in,    // [B,N,8]
                   float* __restrict__ out)           // [B,C,N]
{
    __shared__ float rows[CH * S_];   // 64 KB
    const int b  = blockIdx.y;
    const int c0 = blockIdx.x * CH;
    const int t  = threadIdx.x;

    const float* src = &sf[((size_t)b * C_ + c0) * S_];
    // async memory->LDS staging: 16 x b128 per lane, no VGPR round-trip
    for (int j = t * 4; j < CH * S_; j += 256 * 4) {
        __attribute__((address_space(3))) float* lp =
            (__attribute__((address_space(3))) float*)&rows[j];
        asm volatile("global_load_async_to_lds_b128 %0, %1, off"
                     :: "v"(lp), "v"(src + j) : "memory");
    }
    asm volatile("s_wait_asynccnt 0x0" ::: "memory");
    __syncthreads();

    const int nspan = N_ / NSPLIT;
    const int nbeg  = blockIdx.z * nspan;
    for (int n = nbeg + t; n < nbeg + nspan; n += 256) {
        const size_t base = ((size_t)b * N_ + n) * K_;
        __builtin_prefetch(&idx_in[base + 8 * 256], 0, 0);  // global_prefetch_b8
        int4   i0 = *reinterpret_cast<const int4*>(&idx_in[base]);
        int4   i1 = *reinterpret_cast<const int4*>(&idx_in[base + 4]);
        float4 w0 = *reinterpret_cast<const float4*>(&w_in[base]);
        float4 w1 = *reinterpret_cast<const float4*>(&w_in[base + 4]);
        int   id[8] = {i0.x, i0.y, i0.z, i0.w, i1.x, i1.y, i1.z, i1.w};
        float w [8] = {w0.x, w0.y, w0.z, w0.w, w1.x, w1.y, w1.z, w1.w};
        #pragma unroll
        for (int c = 0; c < CH; ++c) {
            float acc = 0.0f;
            #pragma unroll
            for (int k = 0; k < K_; ++k) acc += w[k] * rows[c * S_ + id[k]];
            out[((size_t)b * C_ + (c0 + c)) * N_ + n] = acc;  // coalesced in n
        }
    }
}

extern "C" void kernel_launch(void* const* d_in, const int* in_sizes, int n_in,
                              void* d_out, int out_size, void* d_ws, size_t ws_size,
                              hipStream_t stream) {
    const float* xyz  = (const float*)d_in[0];   // [B,3,N]
    const float* sxyz = (const float*)d_in[1];   // [B,3,S]
    const float* sf   = (const float*)d_in[2];   // [B,C,S]
    float* out = (float*)d_out;                  // [B,C,N]

    // workspace: [B,N,8] ints (2 MB) then [B,N,8] floats (2 MB)
    int*   idxb = (int*)d_ws;
    float* wb   = (float*)((char*)d_ws + (size_t)B_ * N_ * K_ * sizeof(int));

    knn_weights_kernel<<<dim3(N_ / 128, B_), 256, 0, stream>>>(xyz, sxyz, idxb, wb);
    interp_kernel<<<dim3(C_ / CH, B_, NSPLIT), 256, 0, stream>>>(sf, idxb, wb, out);
}